// DumbMCMC_14714557956383
// MI455X (gfx1250) — compile-verified
//
#include <hip/hip_runtime.h>
#include <math.h>
#include <limits.h>

#define NW 64
#define WAVES_PER_BLOCK 8

// ---------- CDNA5 async global->LDS helpers (gfx1250) ----------
__device__ __forceinline__ void async_load_b128(unsigned lds_addr, const void* gsrc) {
    // GLOBAL_LOAD_ASYNC_TO_LDS_B128: VDST = LDS byte address (vgpr), VADDR = 64-bit global addr
    asm volatile("global_load_async_to_lds_b128 %0, %1, off"
                 :: "v"(lds_addr), "v"(gsrc)
                 : "memory");
}
__device__ __forceinline__ void wait_async0() {
    asm volatile("s_wait_asynccnt 0" ::: "memory");
}
// Low 32 bits of a generic pointer to __shared__ memory are the LDS byte address
__device__ __forceinline__ unsigned lds_addr_of(const void* p) {
    return (unsigned)(unsigned long long)p;
}

__device__ __forceinline__ float clip_u(float x) {
    return fminf(fmaxf(x, 1e-6f), 1.0f - 1e-6f);
}

// Stable rank of the two elements (lane, lane+32) owned by this lane.
// Comparator: (x[k] < x[j]) || (x[k]==x[j] && k<j)  -> jnp stable argsort semantics.
__device__ __forceinline__ void rank64(float x0, float x1, int lane, int& r0, int& r1) {
    r0 = 0; r1 = 0;
#pragma unroll
    for (int k = 0; k < 32; ++k) {
        float xa = __int_as_float(__builtin_amdgcn_readlane(__float_as_int(x0), k));
        float xb = __int_as_float(__builtin_amdgcn_readlane(__float_as_int(x1), k));
        // element j0 = lane
        r0 += (int)((xa < x0) || ((xa == x0) && (k < lane))); // vs element k   (k<32)
        r0 += (int)(xb < x0);                                 // vs element k+32 (tie never counts: k+32 > lane)
        // element j1 = lane+32
        r1 += (int)(xa <= x1);                                // vs element k   (tie always counts: k < lane+32)
        r1 += (int)((xb < x1) || ((xb == x1) && (k < lane))); // vs element k+32
    }
}

// ---------------- Kernel 1: per-row permutation score ----------------
__global__ void __launch_bounds__(256)
score_kernel(const float* __restrict__ rand_u, const float* __restrict__ u,
             const float* __restrict__ bigram, const float* __restrict__ startv,
             const float* __restrict__ endv, float* __restrict__ w_out,
             float* __restrict__ t_out, int chain_size) {
    __shared__ __align__(16) float sB[NW * NW];  // 16 KB bigram
    __shared__ float sS[NW];
    __shared__ float sE[NW];
    __shared__ int   srp[WAVES_PER_BLOCK][NW];

    const int tid  = threadIdx.x;
    const int lane = tid & 31;
    const int wv   = tid >> 5;

    // Stage bigram via async DMA (4 issues x 256 lanes x 16B = 16 KB)
#pragma unroll
    for (int it = 0; it < 4; ++it) {
        int f = (it * 256 + tid) * 4;  // float index, 16B aligned
        async_load_b128(lds_addr_of(&sB[f]), bigram + f);
    }
    if (tid < NW)            sS[tid]      = startv[tid];
    else if (tid < 2 * NW)   sE[tid - NW] = endv[tid - NW];
    wait_async0();
    __syncthreads();

    const int wgid    = blockIdx.x * WAVES_PER_BLOCK + wv;
    const int wstride = gridDim.x * WAVES_PER_BLOCK;

    for (int row = wgid; row < chain_size; row += wstride) {
        const float* rowp = rand_u + (size_t)row * NW;
        if (row + wstride < chain_size)
            __builtin_prefetch(rand_u + (size_t)(row + wstride) * NW, 0, 1);

        float x0 = clip_u(rowp[lane]);
        float x1 = clip_u(rowp[lane + 32]);

        int r0, r1;
        rank64(x0, x1, lane, r0, r1);

        int ng = (int)((r0 != lane) || (r1 != lane + 32)); // per-lane non-gold evidence

        // scatter rp[rank] = index (ranks form a permutation of 0..63)
        srp[wv][r0] = lane;
        srp[wv][r1] = lane + 32;
        asm volatile("" ::: "memory"); // DS ops are in-order within a wave

        // transitions: lane covers r=lane (0..31) and r=lane+32 (32..62)
        int a0 = srp[wv][lane];
        int b0 = srp[wv][lane + 1];
        float part = sB[a0 * NW + b0];
        int r2 = lane + 32;
        if (r2 < NW - 1) {
            int a1 = srp[wv][r2];
            int b1 = srp[wv][r2 + 1];
            part += sB[a1 * NW + b1];
        }
        if (lane == 0)
            part += sS[srp[wv][0]] + sE[srp[wv][NW - 1]];

#pragma unroll
        for (int m = 16; m >= 1; m >>= 1) {
            part += __shfl_xor(part, m, 32);
            ng   |= __shfl_xor(ng, m, 32);
        }
        if (lane == 0) {
            w_out[row] = part;
            // gold rows can never be accepted -> threshold +inf
            t_out[row] = ng ? logf(u[row]) : __builtin_inff();
        }
    }
}

// ---------------- Kernel 2: sequential MH scan (single wave) ----------------
__global__ void scan_kernel(const float* __restrict__ w, const float* __restrict__ t,
                            int* __restrict__ a_out, int chain_size,
                            const int* __restrict__ N_ptr, int n_sel) {
    const int lane = threadIdx.x & 31;
    const int N = *N_ptr;
    const float INF = __builtin_inff();

    // idx0 = first non-gold row
    int idx0 = INT_MAX;
    for (int base = 0; base < chain_size; base += 32) {
        int i = base + lane;
        int cand = (i < chain_size && t[i] != INF) ? i : INT_MAX;
#pragma unroll
        for (int m = 16; m >= 1; m >>= 1)
            cand = min(cand, __shfl_xor(cand, m, 32));
        if (cand != INT_MAX) { idx0 = cand; break; }
    }
    if (idx0 == INT_MAX) idx0 = 0;

    int   cur    = idx0;
    float w_last = w[0];   // reference sets w_last = w[0] regardless of idx0
    int   si  = 0;
    int   sel = N - 1;
    if (sel == 0) { if (lane == 0 && si < n_sel) a_out[si] = idx0; si++; sel += N; }

    for (int base = 1; base < chain_size; base += 32) {
        int i = base + lane;
        float wl = (i < chain_size) ? w[i] : 0.0f;
        float tl = (i < chain_size) ? t[i] : INF;
#pragma unroll
        for (int j = 0; j < 32; ++j) {
            int gi = base + j;
            if (gi >= chain_size) break; // uniform across lanes
            float wj = __int_as_float(__builtin_amdgcn_readlane(__float_as_int(wl), j));
            float tj = __int_as_float(__builtin_amdgcn_readlane(__float_as_int(tl), j));
            if (wj - w_last > tj) { cur = gi; w_last = wj; } // accept
            if (gi == sel) {
                if (lane == 0 && si < n_sel) a_out[si] = cur;
                si++; sel += N;
            }
        }
    }
    while (si < n_sel) { if (lane == 0) a_out[si] = cur; si++; } // jnp OOB-gather clamp
}

// ---------------- Kernel 3: re-rank selected rows, emit permutations ----------------
__global__ void __launch_bounds__(256)
emit_kernel(const float* __restrict__ rand_u, const int* __restrict__ a_sel,
            float* __restrict__ out, int n_sel) {
    const int lane = threadIdx.x & 31;
    const int wv   = threadIdx.x >> 5;
    const int s    = blockIdx.x * WAVES_PER_BLOCK + wv;
    if (s >= n_sel) return;

    const int row = a_sel[s];
    const float* rowp = rand_u + (size_t)row * NW;
    float x0 = clip_u(rowp[lane]);
    float x1 = clip_u(rowp[lane + 32]);

    int r0, r1;
    rank64(x0, x1, lane, r0, r1);

    float* op = out + (size_t)s * NW;   // rp[rank] = element index
    op[r0] = (float)lane;
    op[r1] = (float)(lane + 32);
}

extern "C" void kernel_launch(void* const* d_in, const int* in_sizes, int n_in,
                              void* d_out, int out_size, void* d_ws, size_t ws_size,
                              hipStream_t stream) {
    const float* rand_u = (const float*)d_in[0];
    const float* u      = (const float*)d_in[1];
    const float* bigram = (const float*)d_in[2];
    const float* startv = (const float*)d_in[3];
    const float* endv   = (const float*)d_in[4];
    const int*   N_ptr  = (const int*)d_in[6];

    const int chain_size = in_sizes[1];          // |u|
    const int n_words    = in_sizes[3];          // |start| == 64
    const int n_sel      = out_size / n_words;   // 65536

    float* w_buf = (float*)d_ws;
    float* t_buf = w_buf + chain_size;
    int*   a_buf = (int*)(t_buf + chain_size);

    const int blocks1 = 8192; // grid-stride, 8 waves/block, 1 row/wave/iter
    score_kernel<<<blocks1, 256, 0, stream>>>(rand_u, u, bigram, startv, endv,
                                              w_buf, t_buf, chain_size);

    scan_kernel<<<1, 32, 0, stream>>>(w_buf, t_buf, a_buf, chain_size, N_ptr, n_sel);

    const int blocks3 = (n_sel + WAVES_PER_BLOCK - 1) / WAVES_PER_BLOCK;
    emit_kernel<<<blocks3, 256, 0, stream>>>(rand_u, a_buf, (float*)d_out, n_sel);
}